// Gate_37263136260194
// MI455X (gfx1250) — compile-verified
//
#include <hip/hip_runtime.h>
#include <hip/hip_bf16.h>
#include <math.h>

typedef float v2f __attribute__((ext_vector_type(2)));
typedef float v8f __attribute__((ext_vector_type(8)));

#define N_TOKENS   16384
#define DIM        4096
#define N_EXPERTS  64
#define TOPK       8

#define THREADS        128          // 4 wave32s
#define TOK_PER_BLOCK  64           // 16 tokens per wave
#define KB             32           // K-chunk staged in LDS
#define LSTR           36           // padded LDS row stride: bank = 4*(9*row%16)+col, conflict-free
#define LSTR_L         65           // padded logits row stride
#define BUF_FLOATS     ((N_EXPERTS + TOK_PER_BLOCK) * LSTR)   // 4608 floats = 18 KB / buffer
#define NCHUNK         (DIM / KB)   // 128

// Async global -> LDS copy, 16 bytes per active lane (GV mode, ASYNCcnt-tracked).
__device__ __forceinline__ void async_copy16(unsigned int ldsByteAddr, const float* gsrc) {
    asm volatile("global_load_async_to_lds_b128 %0, %1, off"
                 :: "v"(ldsByteAddr), "v"(gsrc)
                 : "memory");
}
__device__ __forceinline__ void wait_async0() {
    asm volatile("s_wait_asynccnt 0x0" ::: "memory");
}

// Stage one K-chunk (W: 64xKB, x: 64xKB) into the LDS buffer at ldsBase.
// 512+512 float4 tiles / 128 threads = 8 async issues per thread, fully coalesced.
__device__ __forceinline__ void stage_chunk(const float* __restrict__ x,
                                            const float* __restrict__ W,
                                            int tokBlock, int k0,
                                            unsigned int ldsBase, int tid) {
    #pragma unroll
    for (int i = 0; i < (N_EXPERTS * KB / 4) / THREADS; ++i) {   // W chunk
        int q = tid + i * THREADS;          // float4 index
        int e = q >> 3;                     // KB/4 = 8 float4 per expert row
        int c = (q & 7) << 2;
        async_copy16(ldsBase + ((unsigned)(e * LSTR + c) << 2),
                     &W[(size_t)e * DIM + k0 + c]);
    }
    #pragma unroll
    for (int i = 0; i < (TOK_PER_BLOCK * KB / 4) / THREADS; ++i) {  // x tile
        int q = tid + i * THREADS;
        int r = q >> 3;
        int c = (q & 7) << 2;
        async_copy16(ldsBase + ((unsigned)((N_EXPERTS + r) * LSTR + c) << 2),
                     &x[(size_t)(tokBlock + r) * DIM + k0 + c]);
    }
}

__global__ __launch_bounds__(THREADS)
void moe_gate_wmma(const float* __restrict__ x,
                   const float* __restrict__ W,
                   const float* __restrict__ b,
                   float* __restrict__ out)
{
    __shared__ float smem[2 * BUF_FLOATS];   // 36 KB, double-buffered; buf0 reused for logits
    float* sL = smem;                        // logits [64][LSTR_L]

    const int tid  = threadIdx.x;
    const int wave = tid >> 5;
    const int lane = tid & 31;
    const int half = lane >> 4;     // K sub-tile select (WMMA f32 A/B layout)
    const int m16  = lane & 15;     // row (A) / column (B) within 16
    const int tokBlock = blockIdx.x * TOK_PER_BLOCK;

    // LDS byte offsets of the two buffers (low 32 bits of generic shared address).
    const unsigned int buf0Addr = (unsigned int)(size_t)smem;
    const unsigned int buf1Addr = buf0Addr + BUF_FLOATS * 4;

    v8f acc0 = {}, acc1 = {}, acc2 = {}, acc3 = {};

    // Prologue: stage chunk 0 into buffer 0.
    stage_chunk(x, W, tokBlock, 0, buf0Addr, tid);
    wait_async0();
    __syncthreads();

    for (int ch = 0; ch < NCHUNK; ++ch) {
        // Kick off async staging of the next chunk into the other buffer;
        // the async engine fills it while we run WMMA on the current buffer.
        if (ch + 1 < NCHUNK) {
            unsigned int nextAddr = (ch & 1) ? buf0Addr : buf1Addr;
            stage_chunk(x, W, tokBlock, (ch + 1) * KB, nextAddr, tid);
        }

        const float* buf   = smem + (ch & 1) * BUF_FLOATS;
        const float* aBase = buf + N_EXPERTS * LSTR + (wave * 16 + m16) * LSTR + 2 * half;
        const float* bBase = buf + m16 * LSTR + 2 * half;
        #pragma unroll
        for (int kk = 0; kk < KB; kk += 4) {
            v2f a  = *(const v2f*)(aBase + kk);
            v2f b0 = *(const v2f*)(bBase + kk);
            v2f b1 = *(const v2f*)(bBase + 16 * LSTR + kk);
            v2f b2 = *(const v2f*)(bBase + 32 * LSTR + kk);
            v2f b3 = *(const v2f*)(bBase + 48 * LSTR + kk);
            acc0 = __builtin_amdgcn_wmma_f32_16x16x4_f32(false, a, false, b0, (short)0, acc0, false, false);
            acc1 = __builtin_amdgcn_wmma_f32_16x16x4_f32(false, a, false, b1, (short)0, acc1, false, false);
            acc2 = __builtin_amdgcn_wmma_f32_16x16x4_f32(false, a, false, b2, (short)0, acc2, false, false);
            acc3 = __builtin_amdgcn_wmma_f32_16x16x4_f32(false, a, false, b3, (short)0, acc3, false, false);
        }

        wait_async0();      // my async writes to the next buffer are complete
        __syncthreads();    // everyone's are; safe to read next buffer / overwrite this one
    }

    // ---- scatter logits (+bias) to LDS per C-matrix layout ----
    // C tile: lane%16 = expert col, VGPR v = token row v (lanes 0-15) / v+8 (lanes 16-31)
    float bs0 = b[m16], bs1 = b[16 + m16], bs2 = b[32 + m16], bs3 = b[48 + m16];
    #pragma unroll
    for (int v = 0; v < 8; ++v) {
        int tokL = wave * 16 + v + 8 * half;
        sL[tokL * LSTR_L + m16]      = acc0[v] + bs0;
        sL[tokL * LSTR_L + 16 + m16] = acc1[v] + bs1;
        sL[tokL * LSTR_L + 32 + m16] = acc2[v] + bs2;
        sL[tokL * LSTR_L + 48 + m16] = acc3[v] + bs3;
    }
    __syncthreads();

    // ---- per-token top-8 + renormalized softmax (softmax denominator cancels) ----
    if (tid < TOK_PER_BLOCK) {
        const float* row = &sL[tid * LSTR_L];
        unsigned long long used = 0ull;
        float lv[TOPK];
        int   li[TOPK];
        for (int j = 0; j < TOPK; ++j) {
            float best = -INFINITY;
            int   bi   = 0;
            for (int e = 0; e < N_EXPERTS; ++e) {
                if (!((used >> e) & 1ull)) {
                    float v = row[e];
                    if (v > best) { best = v; bi = e; }  // strict > : lowest index wins ties
                }
            }
            used |= 1ull << bi;
            lv[j] = best;
            li[j] = bi;
        }
        float m = lv[0];
        float w[TOPK];
        float s = 0.0f;
        #pragma unroll
        for (int j = 0; j < TOPK; ++j) { w[j] = __expf(lv[j] - m); s += w[j]; }
        float inv = 1.0f / s;

        int tok = tokBlock + tid;
        float* wOut = out + (size_t)tok * TOPK;
        float* iOut = out + (size_t)N_TOKENS * TOPK + (size_t)tok * TOPK;
        #pragma unroll
        for (int j = 0; j < TOPK; ++j) {
            wOut[j] = w[j] * inv;
            iOut[j] = (float)li[j];
        }
    }
}

extern "C" void kernel_launch(void* const* d_in, const int* in_sizes, int n_in,
                              void* d_out, int out_size, void* d_ws, size_t ws_size,
                              hipStream_t stream) {
    const float* x = (const float*)d_in[0];
    const float* W = (const float*)d_in[1];
    const float* b = (const float*)d_in[2];
    float* out = (float*)d_out;
    dim3 grid(N_TOKENS / TOK_PER_BLOCK);   // 256 blocks
    dim3 block(THREADS);                   // 128 threads = 4 wave32
    moe_gate_wmma<<<grid, block, 0, stream>>>(x, W, b, out);
}